// PyramidROIAlign_18013092840122
// MI455X (gfx1250) — compile-verified
//
#include <hip/hip_runtime.h>

// Pyramid ROI Align for MI455X (gfx1250): bilinear crop_and_resize expressed
// as a per-box GEMM  out[49x256] = A[49x196] x G[196x256]  on
// v_wmma_f32_16x16x4_f32 (fp32 in/out -> matches the f32 reference math).
//
// Roofline: ~100MB written + ~400MB gathered reads (196 corner rows x 1KB per
// box, heavy L2 reuse) => ~20us floor at 23.3 TB/s. Compute is trivial, so the
// design goal is minimal per-gather overhead: one ds_load_b64 of a precombined
// byte-offset pair + one coalesced global_load_b32 per B element, software-
// pipelined one k-step ahead of the WMMAs.

typedef __attribute__((ext_vector_type(2))) float v2f;
typedef __attribute__((ext_vector_type(8))) float v8f;

#define POOLM  49        // 7x7 output pixels
#define MPAD   64        // padded M (4 tiles of 16)
#define KTOT   196       // 14x14 corner grid
#define KSTEPS (KTOT/4)  // 49 WMMA k-steps
#define NCH    256       // channels

__global__ __launch_bounds__(256) void roialign_wmma_kernel(
    const float* __restrict__ boxes, const float* __restrict__ meta,
    const float* __restrict__ f2, const float* __restrict__ f3,
    const float* __restrict__ f4, const float* __restrict__ f5,
    float* __restrict__ out, int N)
{
    __shared__ __align__(16) float sA[MPAD * KTOT];  // dense interp matrix [m][k]
    __shared__ uint2 sOff2[KTOT / 2];                // {off[2t], off[2t+1]} global byte offsets
    __shared__ unsigned int sRow[14];                // grid-row byte offset (incl. batch)
    __shared__ unsigned int sCol[14];                // grid-col byte offset
    __shared__ float sLy[7], sLx[7];
    __shared__ int sLevel;

    const int box = blockIdx.x;
    const int b   = box / N;
    const int tid = threadIdx.x;

    if (tid == 0) {
        const float y1 = boxes[box*4 + 0];
        const float x1 = boxes[box*4 + 1];
        const float y2 = boxes[box*4 + 2];
        const float x2 = boxes[box*4 + 3];
        const float h = y2 - y1, w = x2 - x1;
        const float imgH = meta[4], imgW = meta[5];
        const float scale = 224.0f / sqrtf(imgH * imgW);   // 224/sqrt(area)
        const float rl = log2f(sqrtf(h * w) / scale);
        int level = 4 + (int)rintf(rl);                    // jnp.round = RNE
        level = level < 2 ? 2 : (level > 5 ? 5 : level);
        const int li = level - 2;
        sLevel = li;
        const int H = 256 >> li;                           // 256,128,64,32
        const int W = H;

        #pragma unroll
        for (int i = 0; i < 7; ++i) {
            // y grid
            float ys = (y1 + (y2 - y1) * (float)i * (1.0f/6.0f)) * (float)(H - 1);
            float fy = floorf(ys);
            fy = fminf(fmaxf(fy, 0.0f), (float)(H - 1));
            int yy0 = (int)fy;
            int yy1 = yy0 + 1; if (yy1 > H - 1) yy1 = H - 1;
            sLy[i] = ys - fy;
            sRow[2*i    ] = (unsigned)((b * H + yy0) * W) * (NCH * 4u);
            sRow[2*i + 1] = (unsigned)((b * H + yy1) * W) * (NCH * 4u);
            // x grid
            float xs = (x1 + (x2 - x1) * (float)i * (1.0f/6.0f)) * (float)(W - 1);
            float fx = floorf(xs);
            fx = fminf(fmaxf(fx, 0.0f), (float)(W - 1));
            int xx0 = (int)fx;
            int xx1 = xx0 + 1; if (xx1 > W - 1) xx1 = W - 1;
            sLx[i] = xs - fx;
            sCol[2*i    ] = (unsigned)xx0 * (NCH * 4u);
            sCol[2*i + 1] = (unsigned)xx1 * (NCH * 4u);
        }
    }
    __syncthreads();

    // Combined gather-offset pairs: sOff2[t] = {row+col byte offset of k=2t, of k=2t+1}
    for (int t = tid; t < KTOT / 2; t += 256) {
        const int k0 = 2 * t, k1 = k0 + 1;
        const int r0 = k0 / 14, s0 = k0 - 14 * r0;
        const int r1 = k1 / 14, s1 = k1 - 14 * r1;
        sOff2[t] = make_uint2(sRow[r0] + sCol[s0], sRow[r1] + sCol[s1]);
    }

    // Dense A: A[m][k] = wy(i, r) * wx(j, s), m = i*7+j, k = r*14+s
    for (int idx = tid; idx < MPAD * KTOT; idx += 256) {
        const int m = idx / KTOT;
        const int k = idx - m * KTOT;
        float v = 0.0f;
        if (m < POOLM) {
            const int i = m / 7, j = m - 7 * i;
            const int r = k / 14, s = k - 14 * r;
            const float wy = ((r >> 1) == i) ? ((r & 1) ? sLy[i] : 1.0f - sLy[i]) : 0.0f;
            const float wx = ((s >> 1) == j) ? ((s & 1) ? sLx[j] : 1.0f - sLx[j]) : 0.0f;
            v = wy * wx;
        }
        sA[idx] = v;
    }
    __syncthreads();

    const char* base;
    {
        const int li = sLevel;
        base = (const char*)(li == 0 ? f2 : (li == 1 ? f3 : (li == 2 ? f4 : f5)));
    }

    const int wave = tid >> 5;
    const int lane = tid & 31;
    const int hi   = lane >> 4;     // half-wave select: K offset 0 or 2
    const int ln   = lane & 15;

    // Per-M-tile A row pointers (stride 196 dwords: conflict-free, 8B-aligned pairs).
    const float* __restrict__ A0 = &sA[(ln +  0) * KTOT];
    const float* __restrict__ A1 = &sA[(ln + 16) * KTOT];
    const float* __restrict__ A2 = &sA[(ln + 32) * KTOT];
    const float* __restrict__ A3 = &sA[(ln + 48) * KTOT];

    // Each wave owns 2 N-tiles of 16 channels (8 waves x 32 = 256 channels).
    for (int ntl = 0; ntl < 2; ++ntl) {
        const int c0 = (wave * 2 + ntl) * 16;
        const unsigned cByte = (unsigned)(c0 + ln) * 4u;

        v8f acc0 = {}, acc1 = {}, acc2 = {}, acc3 = {};

        // Software pipeline: B values for step kk+1 load while step kk's WMMAs run.
        uint2 oc = sOff2[hi];
        float b0 = *(const float*)(base + (oc.x + cByte));
        float b1 = *(const float*)(base + (oc.y + cByte));

        for (int kk = 0; kk < KSTEPS; ++kk) {
            v2f bf; bf.x = b0; bf.y = b1;

            if (kk + 1 < KSTEPS) {
                const uint2 on = sOff2[(kk + 1) * 2 + hi];
                b0 = *(const float*)(base + (on.x + cByte));
                b1 = *(const float*)(base + (on.y + cByte));
            }

            // A 16x4 layout: VGPR0 = K(kbase + 2*hi), VGPR1 = K+1, M = ln.
            const int k0 = kk * 4 + (hi << 1);
            v2f a0, a1, a2, a3;
            a0.x = A0[k0];  a0.y = A0[k0 + 1];
            a1.x = A1[k0];  a1.y = A1[k0 + 1];
            a2.x = A2[k0];  a2.y = A2[k0 + 1];
            a3.x = A3[k0];  a3.y = A3[k0 + 1];

            acc0 = __builtin_amdgcn_wmma_f32_16x16x4_f32(false, a0, false, bf, (short)0, acc0, false, false);
            acc1 = __builtin_amdgcn_wmma_f32_16x16x4_f32(false, a1, false, bf, (short)0, acc1, false, false);
            acc2 = __builtin_amdgcn_wmma_f32_16x16x4_f32(false, a2, false, bf, (short)0, acc2, false, false);
            acc3 = __builtin_amdgcn_wmma_f32_16x16x4_f32(false, a3, false, bf, (short)0, acc3, false, false);
        }

        // C/D layout: VGPR v holds rows v (lanes 0-15) and v+8 (lanes 16-31), N = ln.
        #define STORE_TILE(ACC, MT)                                                  \
            _Pragma("unroll")                                                        \
            for (int v = 0; v < 8; ++v) {                                            \
                const int m = (MT) * 16 + v + hi * 8;                                \
                if (m < POOLM)                                                       \
                    out[((size_t)box * POOLM + m) * NCH + c0 + ln] = (ACC)[v];       \
            }
        STORE_TILE(acc0, 0)
        STORE_TILE(acc1, 1)
        STORE_TILE(acc2, 2)
        STORE_TILE(acc3, 3)
        #undef STORE_TILE
    }
}

extern "C" void kernel_launch(void* const* d_in, const int* in_sizes, int n_in,
                              void* d_out, int out_size, void* d_ws, size_t ws_size,
                              hipStream_t stream) {
    const float* boxes = (const float*)d_in[0];
    const float* meta  = (const float*)d_in[1];
    const float* f2    = (const float*)d_in[2];
    const float* f3    = (const float*)d_in[3];
    const float* f4    = (const float*)d_in[4];
    const float* f5    = (const float*)d_in[5];

    const int B = in_sizes[1] / 93;          // image_meta is [B, 93]
    const int N = in_sizes[0] / (4 * B);     // boxes is [B, N, 4]

    roialign_wmma_kernel<<<B * N, 256, 0, stream>>>(
        boxes, meta, f2, f3, f4, f5, (float*)d_out, N);
}